// STN_fixTheta_86620900426059
// MI455X (gfx1250) — compile-verified
//
#include <hip/hip_runtime.h>

// ---------------- fixed problem geometry (from setup_inputs) ----------------
#define NB      8
#define SRC_H   128
#define SRC_W   256
#define UP_H    1024          // SRC_H*8 == output_sizey
#define UP_W    2048          // SRC_W*8 == output_sizex
#define OUT_H   1024
#define OUT_W   2048

#define BLOCK           512
#define BLOCKS_PER_B    64
#define PX_PER_THREAD   4
#define ROWS_PER_BLOCK  (OUT_H / BLOCKS_PER_B)   // 16 rows, one row per iteration

typedef __attribute__((ext_vector_type(4))) unsigned int u32x4;
typedef __attribute__((ext_vector_type(8))) int          i32x8;
typedef __attribute__((ext_vector_type(4))) int          i32x4;
typedef __attribute__((ext_vector_type(4))) float        f32x4;

// Gather one corner of the (virtually) 8x-upsampled image from the LDS tile.
// Upsample is folded into the index math: gt_up[y][x] == tile[y>>3][x>>3].
__device__ __forceinline__ float fetch(const float* __restrict__ t, int x, int y) {
  bool v = ((unsigned)x < (unsigned)UP_W) && ((unsigned)y < (unsigned)UP_H);
  int xs = min(max(x, 0), UP_W - 1) >> 3;
  int ys = min(max(y, 0), UP_H - 1) >> 3;
  float val = t[(ys << 8) + xs];          // SRC_W == 256
  return v ? val : 0.0f;
}

// torch grid_sample: bilinear, zeros padding, align_corners=False
__device__ __forceinline__ float bsample(const float* __restrict__ tile, float gx, float gy) {
  float ix = ((gx + 1.0f) * (float)UP_W - 1.0f) * 0.5f;
  float iy = ((gy + 1.0f) * (float)UP_H - 1.0f) * 0.5f;
  float fx0 = floorf(ix), fy0 = floorf(iy);
  float wx = ix - fx0,    wy = iy - fy0;
  int x0 = (int)fx0, y0 = (int)fy0;
  float v00 = fetch(tile, x0,     y0);
  float v10 = fetch(tile, x0 + 1, y0);
  float v01 = fetch(tile, x0,     y0 + 1);
  float v11 = fetch(tile, x0 + 1, y0 + 1);
  return v00 * (1.0f - wx) * (1.0f - wy) + v10 * wx * (1.0f - wy)
       + v01 * (1.0f - wx) * wy          + v11 * wx * wy;
}

__global__ void __launch_bounds__(BLOCK)
stn_kernel(const float* __restrict__ gt, const float* __restrict__ theta,
           float* __restrict__ out) {
  extern __shared__ float tile[];   // SRC_H*SRC_W f32 = 128 KB (fits CDNA5 320KB LDS)
  const int b   = blockIdx.x / BLOCKS_PER_B;
  const int blk = blockIdx.x % BLOCKS_PER_B;

  // ---- stage gt[b] (128x256 f32, 128 KB) into LDS via the Tensor Data Mover ----
#if __has_builtin(__builtin_amdgcn_tensor_load_to_lds) && __has_builtin(__builtin_amdgcn_s_wait_tensorcnt)
  if (threadIdx.x < 32u) {          // first wave issues the DMA (EXEC-independent op)
    unsigned long long ga = (unsigned long long)(const void*)(gt + (size_t)b * (SRC_H * SRC_W));
    unsigned lds_off = (unsigned)(unsigned long long)(const void*)tile;  // low 32b = LDS offset
    // D# group 0: count=1 valid, lds_addr, 57b global_addr, type=2 ("image")
    u32x4 g0;
    g0[0] = 1u;
    g0[1] = lds_off;
    g0[2] = (unsigned)ga;
    g0[3] = ((unsigned)(ga >> 32) & 0x01FFFFFFu) | 0x80000000u;
    // D# group 1: no multicast, data_size=4B, dims/tile dims/strides (element units)
    i32x8 g1;
    g1[0] = (int)(2u << 16);                          // data_size code 2 -> 4 bytes
    g1[1] = (int)((unsigned)SRC_W << 16);             // tensor_dim0[15:0] @ bits[63:48]
    g1[2] = (int)((unsigned)SRC_H << 16);             // tensor_dim1[15:0] @ bits[95:80]
    g1[3] = (int)((unsigned)SRC_W << 16);             // tile_dim0 @ bits[127:112]
    g1[4] = (int)SRC_H;                               // tile_dim1 @ bits[143:128], tile_dim2=0
    g1[5] = (int)SRC_W;                               // tensor_dim0_stride lo32
    g1[6] = (int)((unsigned)(SRC_H * SRC_W) << 16);   // tensor_dim1_stride[15:0] @ [223:208]
    g1[7] = (int)((unsigned)(SRC_H * SRC_W) >> 16);   // tensor_dim1_stride hi
    i32x4 g2 = {0, 0, 0, 0};                          // 2-D tensor: groups 2/3 -> NULL
    i32x4 g3 = {0, 0, 0, 0};
    i32x8 g4 = {0, 0, 0, 0, 0, 0, 0, 0};              // trailing group (clang-23 6-arg form)
    __builtin_amdgcn_tensor_load_to_lds(g0, g1, g2, g3, g4, 0);
    __builtin_amdgcn_s_wait_tensorcnt(0);             // data resident before barrier
  }
#else
  {
    const float* src = gt + (size_t)b * (SRC_H * SRC_W);
    for (int k = threadIdx.x; k < SRC_H * SRC_W; k += BLOCK) tile[k] = src[k];
  }
#endif
  __syncthreads();

  const float t0 = theta[b * 6 + 0], t1 = theta[b * 6 + 1], t2 = theta[b * 6 + 2];
  const float t3 = theta[b * 6 + 3], t4 = theta[b * 6 + 4], t5 = theta[b * 6 + 5];

  float* __restrict__ outy = out + (size_t)NB * OUT_H * OUT_W;   // boxy follows box
  const int wo = threadIdx.x * PX_PER_THREAD;

  for (int i = 0; i < ROWS_PER_BLOCK; ++i) {
    const int ho = blk * ROWS_PER_BLOCK + i;
    const float yn = ((float)ho + 0.5f) * (2.0f / OUT_H) - 1.0f;

    // boxy: thetay = [[1,0,0],[0,t4,t5]] => grid-x == xn exactly, so the x-lerp
    // weight is exactly 0 and the y-interp parameters are wave-uniform (SALU).
    const float gy2 = t4 * yn + t5;
    const float iy2 = ((gy2 + 1.0f) * (float)UP_H - 1.0f) * 0.5f;
    const float fy2 = floorf(iy2);
    const float wy2 = iy2 - fy2;
    const int   y0b = (int)fy2;
    const bool  vy0 = ((unsigned)y0b       < (unsigned)UP_H);
    const bool  vy1 = ((unsigned)(y0b + 1) < (unsigned)UP_H);
    const int   ys0 = (min(max(y0b,     0), UP_H - 1) >> 3) << 8;
    const int   ys1 = (min(max(y0b + 1, 0), UP_H - 1) >> 3) << 8;

    float r[PX_PER_THREAD], ry[PX_PER_THREAD];
#pragma unroll
    for (int j = 0; j < PX_PER_THREAD; ++j) {
      const float xn = ((float)(wo + j) + 0.5f) * (2.0f / OUT_W) - 1.0f;
      // box: full affine bilinear sample
      const float gx = t0 * xn + t1 * yn + t2;
      const float gy = t3 * xn + t4 * yn + t5;
      r[j] = bsample(tile, gx, gy);
      // boxy: degenerate x-lerp, pure y interpolation at column wo+j
      const int xs = (wo + j) >> 3;
      const float a = vy0 ? tile[ys0 + xs] : 0.0f;
      const float c = vy1 ? tile[ys1 + xs] : 0.0f;
      ry[j] = a * (1.0f - wy2) + c * wy2;
    }

    const size_t o = (size_t)b * (OUT_H * OUT_W) + (size_t)ho * OUT_W + wo;
    // Streaming outputs: non-temporal b128 stores (write-once, never re-read here).
    f32x4 v0 = { r[0],  r[1],  r[2],  r[3]  };
    f32x4 v1 = { ry[0], ry[1], ry[2], ry[3] };
    __builtin_nontemporal_store(v0, (f32x4*)(out  + o));
    __builtin_nontemporal_store(v1, (f32x4*)(outy + o));
  }
}

extern "C" void kernel_launch(void* const* d_in, const int* in_sizes, int n_in,
                              void* d_out, int out_size, void* d_ws, size_t ws_size,
                              hipStream_t stream) {
  (void)in_sizes; (void)n_in; (void)out_size; (void)d_ws; (void)ws_size;
  const float* gt    = (const float*)d_in[0];   // [8,1,128,256] f32
  const float* theta = (const float*)d_in[1];   // [8,6] f32
  dim3 grid(NB * BLOCKS_PER_B);                 // 512 workgroups
  dim3 block(BLOCK);                            // 512 threads = 16 wave32
  size_t shmem = (size_t)SRC_H * SRC_W * sizeof(float);   // 128 KB dynamic LDS
  hipLaunchKernelGGL(stn_kernel, grid, block, shmem, stream,
                     gt, theta, (float*)d_out);
}